// SBLocalAttention_25391846654025
// MI455X (gfx1250) — compile-verified
//
#include <hip/hip_runtime.h>
#include <math.h>
#include <stdint.h>

// ---------------- problem constants (B=1) ----------------
#define T_DIM   1024      // sequence length (queries == keys)
#define H_DIM   8         // heads
#define E_DIM   64        // head dim
#define R_DIM   64        // projection rows
#define F_DIM   128       // 2R feature dim
#define HALF_W  64        // local window half-width (LOCAL_CONTEXT/2)
#define NTILES_T (T_DIM/16)

#define TEMP_SCALE 0.125f                 // 1/sqrt(E)
#define DATA_NORM  0.35355339059327373f   // sqrt(1/sqrt(E))
#define RATIO      0.08838834764831843f   // (2R)^-0.5

typedef float v2f __attribute__((ext_vector_type(2)));
typedef float v8f __attribute__((ext_vector_type(8)));

// CDNA5 fp32 matrix core: D(16x16) = A(16x4) * B(4x16) + C
__device__ __forceinline__ v8f wmma4(v2f a, v2f b, v8f c) {
  return __builtin_amdgcn_wmma_f32_16x16x4_f32(false, a, false, b, (short)0, c,
                                               false, false);
}

__device__ __forceinline__ int clampi(int x, int lo, int hi) {
  return x < lo ? lo : (x > hi ? hi : x);
}

// ---- CDNA5 async global->LDS staging (ASYNCcnt path), guarded ----
#if defined(__has_builtin)
# if __has_builtin(__builtin_amdgcn_global_load_async_to_lds_b128) && \
     __has_builtin(__builtin_amdgcn_s_wait_asynccnt)
#  define USE_ASYNC_LDS 1
# endif
#endif
#ifndef USE_ASYNC_LDS
# define USE_ASYNC_LDS 0
#endif

#if USE_ASYNC_LDS
// builtin signature (per hipcc diagnostic): pointee is a 16-byte int vector
typedef int v4i_t __attribute__((vector_size(16)));
typedef __attribute__((address_space(1))) v4i_t* gas1_t;
typedef __attribute__((address_space(3))) v4i_t* las3_t;
// copy 16 bytes per active lane, global -> LDS, no VGPR round trip
__device__ __forceinline__ void async_copy16(const void* g, const void* l) {
  __builtin_amdgcn_global_load_async_to_lds_b128(
      (gas1_t)(uintptr_t)g, (las3_t)(uint32_t)(uintptr_t)l, 0, 0);
}
__device__ __forceinline__ void async_wait0() {
  __builtin_amdgcn_s_wait_asynccnt(0);
}
#endif

// ---------------------------------------------------------------------------
// K1: featurize q and k tiles.  dash = DATA_NORM * (X @ P^T) via WMMA.
// which==0 (query): emit q_prime + per-row stabilizer q_ls
// which==1 (key)  : stash dash, diag, per-row max (for global k stabilizer)
// ---------------------------------------------------------------------------
__global__ __launch_bounds__(32)
void featurize_kernel(const float* __restrict__ qin, const float* __restrict__ kin,
                      const float* __restrict__ proj,
                      float* __restrict__ qp, float* __restrict__ qls,
                      float* __restrict__ kdash, float* __restrict__ kdiag,
                      float* __restrict__ krowmax)
{
  __shared__ float Pl[R_DIM][E_DIM];
  __shared__ float Xs[16][E_DIM];     // raw (unscaled) data tile
  __shared__ float Dash[16][R_DIM];
  __shared__ float DiagL[16];

  const int lane  = threadIdx.x;
  const int bid   = blockIdx.x;
  const int which = bid / (H_DIM * NTILES_T);
  const int rem   = bid % (H_DIM * NTILES_T);
  const int h     = rem / NTILES_T;
  const int tt    = rem % NTILES_T;
  const float* __restrict__ xin = (which == 0) ? qin : kin;

#if USE_ASYNC_LDS
  for (int c = lane; c < (R_DIM * E_DIM) / 4; c += 32)
    async_copy16(proj + c * 4, &Pl[0][0] + c * 4);
  for (int c = lane; c < (16 * E_DIM) / 4; c += 32) {
    int m = c >> 4, e4 = c & 15;
    int t = tt * 16 + m;
    async_copy16(xin + ((size_t)t * H_DIM + h) * E_DIM + e4 * 4, &Xs[m][e4 * 4]);
  }
  async_wait0();
#else
  for (int i = lane; i < R_DIM * E_DIM; i += 32)
    Pl[i / E_DIM][i % E_DIM] = proj[i];
  for (int i = lane; i < 16 * E_DIM; i += 32) {
    int m = i >> 6, e = i & 63;
    int t = tt * 16 + m;
    Xs[m][e] = xin[((size_t)t * H_DIM + h) * E_DIM + e];
  }
#endif
  __syncthreads();

  if (lane < 16) {
    float s = 0.f;
    for (int e = 0; e < E_DIM; ++e) { float x = Xs[lane][e]; s += x * x; }
    DiagL[lane] = 0.5f * DATA_NORM * DATA_NORM * s;  // 0.5*dn^2*||x||^2
  }

  const int n    = lane & 15;
  const int koff = (lane >> 4) * 2;
  for (int nt = 0; nt < 4; ++nt) {
    v8f acc = {};
    for (int kk = 0; kk < E_DIM / 4; ++kk) {
      int k0 = kk * 4 + koff;
      v2f a, b;
      a.x = Xs[n][k0];            a.y = Xs[n][k0 + 1];
      int r = nt * 16 + n;        // B[k][n] = P^T[k][r] = P[r][k]
      b.x = Pl[r][k0];            b.y = Pl[r][k0 + 1];
      acc = wmma4(a, b, acc);
    }
    for (int vv = 0; vv < 8; ++vv) {
      int row = vv + ((lane >> 4) << 3);
      Dash[row][nt * 16 + n] = DATA_NORM * acc[vv];   // dash = dn * (X @ P^T)
    }
  }
  __syncthreads();

  if (lane < 16) {
    const int t    = tt * 16 + lane;
    const float dg = DiagL[lane];
    float mx = 0.f;                               // max_r |dash_r|  (>=0)
    for (int r = 0; r < R_DIM; ++r) mx = fmaxf(mx, fabsf(Dash[lane][r]));
    if (which == 0) {
      // stab = mx - diag;  prime = ratio*exp(±dash - diag - stab) = ratio*exp(±dash - mx)
      float* qrow = qp + ((size_t)h * T_DIM + t) * F_DIM;
      for (int r = 0; r < R_DIM; ++r) {
        float d = Dash[lane][r];
        qrow[r]         = RATIO * expf( d - mx);
        qrow[r + R_DIM] = RATIO * expf(-d - mx);
      }
      qls[h * T_DIM + t] = mx - dg;
    } else {
      float* krow = kdash + ((size_t)h * T_DIM + t) * R_DIM;
      for (int r = 0; r < R_DIM; ++r) krow[r] = Dash[lane][r];
      kdiag[h * T_DIM + t]   = dg;
      krowmax[h * T_DIM + t] = mx - dg;
    }
  }
}

// ---------------------------------------------------------------------------
// K2: per-head global key stabilizer (deterministic tree reduction)
// ---------------------------------------------------------------------------
__global__ __launch_bounds__(256)
void reduce_kls_kernel(const float* __restrict__ krowmax, float* __restrict__ kls)
{
  __shared__ float sm[256];
  const int h = blockIdx.x, tid = threadIdx.x;
  float m = -INFINITY;
  for (int t = tid; t < T_DIM; t += 256) m = fmaxf(m, krowmax[h * T_DIM + t]);
  sm[tid] = m; __syncthreads();
  for (int s = 128; s > 0; s >>= 1) {
    if (tid < s) sm[tid] = fmaxf(sm[tid], sm[tid + s]);
    __syncthreads();
  }
  if (tid == 0) kls[h] = sm[0];
}

// ---------------------------------------------------------------------------
// K3: k_prime = ratio * exp(±dash - diag - k_ls)
// ---------------------------------------------------------------------------
__global__ __launch_bounds__(256)
void kprime_kernel(const float* __restrict__ kdash, const float* __restrict__ kdiag,
                   const float* __restrict__ kls, float* __restrict__ kp)
{
  int idx = blockIdx.x * 256 + threadIdx.x;       // over H*T*R
  if (idx >= H_DIM * T_DIM * R_DIM) return;
  const int r  = idx % R_DIM;
  const int ht = idx / R_DIM;
  const int h  = ht / T_DIM;
  const float dash = kdash[idx];
  const float c    = kdiag[ht] + kls[h];
  kp[(size_t)ht * F_DIM + r]         = RATIO * expf( dash - c);
  kp[(size_t)ht * F_DIM + r + R_DIM] = RATIO * expf(-dash - c);
}

// ---------------------------------------------------------------------------
// K4: kv[h] = k_prime^T @ V   (F x E, K = T) via WMMA; 1 wave per 16x16 tile
// ---------------------------------------------------------------------------
__global__ __launch_bounds__(32)
void kv_kernel(const float* __restrict__ vin, const float* __restrict__ kp,
               float* __restrict__ kv)
{
  const int bid  = blockIdx.x;
  const int h    = bid / 32;
  const int rem  = bid % 32;
  const int mt   = rem / 4;      // feature tile
  const int nt   = rem % 4;      // d tile
  const int lane = threadIdx.x;
  const int n    = lane & 15;
  const int koff = (lane >> 4) * 2;
  const int mrow = mt * 16 + n;
  const int d    = nt * 16 + n;
  const float* __restrict__ kpb = kp + (size_t)h * T_DIM * F_DIM;

  v8f acc = {};
  for (int kk = 0; kk < T_DIM / 4; ++kk) {
    int s0 = kk * 4 + koff;
    v2f a, b;
    a.x = kpb[(size_t)s0 * F_DIM + mrow];         // A[m][s] = kp[s][m]
    a.y = kpb[(size_t)(s0 + 1) * F_DIM + mrow];
    b.x = vin[((size_t)s0 * H_DIM + h) * E_DIM + d];
    b.y = vin[((size_t)(s0 + 1) * H_DIM + h) * E_DIM + d];
    acc = wmma4(a, b, acc);
  }
  for (int vv = 0; vv < 8; ++vv) {
    int m = mt * 16 + vv + ((lane >> 4) << 3);
    kv[((size_t)h * F_DIM + m) * E_DIM + d] = acc[vv];
  }
}

// ---------------------------------------------------------------------------
// K5: ksum[h][m] = sum_s k_prime[h][s][m]
// ---------------------------------------------------------------------------
__global__ __launch_bounds__(128)
void ksum_kernel(const float* __restrict__ kp, float* __restrict__ ksum)
{
  const int h = blockIdx.x, m = threadIdx.x;
  const float* base = kp + (size_t)h * T_DIM * F_DIM + m;
  float s = 0.f;
  for (int t = 0; t < T_DIM; ++t) s += base[(size_t)t * F_DIM];
  ksum[h * F_DIM + m] = s;
}

// ---------------------------------------------------------------------------
// K6: fused band attention + low-rank combine. 1 wave per (h, 16-query tile).
// 9 aligned key tiles cover the 128-wide window; mask handles edges.
// ---------------------------------------------------------------------------
__global__ __launch_bounds__(32)
void sb_final_kernel(const float* __restrict__ qin, const float* __restrict__ kin,
                     const float* __restrict__ vin,
                     const float* __restrict__ qp, const float* __restrict__ kp,
                     const float* __restrict__ qls, const float* __restrict__ kls,
                     const float* __restrict__ kv, const float* __restrict__ ksum,
                     float* __restrict__ out)
{
  __shared__ float Xq[16][E_DIM];     // raw query tile (temp applied post-WMMA)
  __shared__ float QPl[16][F_DIM];    // q_prime tile
  __shared__ float QKb[16][144];
  __shared__ float DPb[16][144];
  __shared__ float Ab[16][144];
  __shared__ float Pscale[16];

  const int lane   = threadIdx.x;
  const int h      = blockIdx.x / NTILES_T;
  const int qt     = blockIdx.x % NTILES_T;
  const int n      = lane & 15;
  const int koff   = (lane >> 4) * 2;
  const int s_base = (qt - 4) * 16;

#if USE_ASYNC_LDS
  for (int c = lane; c < (16 * E_DIM) / 4; c += 32) {
    int m = c >> 4, e4 = c & 15;
    int t = qt * 16 + m;
    async_copy16(qin + ((size_t)t * H_DIM + h) * E_DIM + e4 * 4, &Xq[m][e4 * 4]);
  }
  for (int c = lane; c < (16 * F_DIM) / 4; c += 32) {
    int m = c >> 5, f4 = c & 31;
    int t = qt * 16 + m;
    async_copy16(qp + ((size_t)h * T_DIM + t) * F_DIM + f4 * 4, &QPl[m][f4 * 4]);
  }
  async_wait0();
#else
  for (int i = lane; i < 16 * E_DIM; i += 32) {
    int m = i >> 6, e = i & 63;
    int t = qt * 16 + m;
    Xq[m][e] = qin[((size_t)t * H_DIM + h) * E_DIM + e];
  }
  for (int i = lane; i < 16 * F_DIM; i += 32) {
    int m = i >> 7, f = i & 127;
    int t = qt * 16 + m;
    QPl[m][f] = qp[((size_t)h * T_DIM + t) * F_DIM + f];
  }
#endif
  __syncthreads();

  // qkv tile: (16 x 128) @ (128 x 64), kept in registers
  v8f qkvAcc[4];
  for (int nt = 0; nt < 4; ++nt) {
    v8f acc = {};
    const int d = nt * 16 + n;
    for (int kk = 0; kk < F_DIM / 4; ++kk) {
      int k0 = kk * 4 + koff;
      v2f a, b;
      a.x = QPl[n][k0];  a.y = QPl[n][k0 + 1];
      b.x = kv[((size_t)h * F_DIM + k0) * E_DIM + d];
      b.y = kv[((size_t)h * F_DIM + k0 + 1) * E_DIM + d];
      acc = wmma4(a, b, acc);
    }
    qkvAcc[nt] = acc;
  }

  // 9 key tiles: QK (K=64) and dots_prime (K=128)
  for (int d8 = 0; d8 < 9; ++d8) {
    const int sc = clampi(s_base + d8 * 16 + n, 0, T_DIM - 1);
    v8f accQ = {};
    for (int kk = 0; kk < E_DIM / 4; ++kk) {
      int k0 = kk * 4 + koff;
      v2f a, b;
      a.x = Xq[n][k0];  a.y = Xq[n][k0 + 1];
      b.x = kin[((size_t)sc * H_DIM + h) * E_DIM + k0];
      b.y = kin[((size_t)sc * H_DIM + h) * E_DIM + k0 + 1];
      accQ = wmma4(a, b, accQ);
    }
    v8f accD = {};
    for (int kk = 0; kk < F_DIM / 4; ++kk) {
      int k0 = kk * 4 + koff;
      v2f a, b;
      a.x = QPl[n][k0];  a.y = QPl[n][k0 + 1];
      b.x = kp[((size_t)h * T_DIM + sc) * F_DIM + k0];
      b.y = kp[((size_t)h * T_DIM + sc) * F_DIM + k0 + 1];
      accD = wmma4(a, b, accD);
    }
    for (int vv = 0; vv < 8; ++vv) {
      int row = vv + ((lane >> 4) << 3);
      QKb[row][d8 * 16 + n] = TEMP_SCALE * accQ[vv];   // QK = temp * (q . k)
      DPb[row][d8 * 16 + n] = accD[vv];
    }
  }
  __syncthreads();

  // per-query-row combine (matches reference logsumexp/clip exactly)
  if (lane < 16) {
    const int row = lane;
    const int t   = qt * 16 + row;
    const float pls = qls[h * T_DIM + t] + kls[h];
    float qk1 = 0.f;
    for (int f = 0; f < F_DIM; ++f) qk1 += QPl[row][f] * ksum[h * F_DIM + f];
    float mx = -INFINITY;
    for (int j = 0; j < 144; ++j) {
      int s = s_base + j;
      bool valid = (s >= 0) && (s < T_DIM) && (s >= t - HALF_W) && (s <= t + HALF_W - 1);
      if (valid) mx = fmaxf(mx, QKb[row][j]);
    }
    float se = 0.f, dpsum = 0.f;
    for (int j = 0; j < 144; ++j) {
      int s = s_base + j;
      bool valid = (s >= 0) && (s < T_DIM) && (s >= t - HALF_W) && (s <= t + HALF_W - 1);
      if (valid) { se += expf(QKb[row][j] - mx); dpsum += DPb[row][j]; }
    }
    const float lse = mx + logf(se);
    const float lr  = logf(fmaxf(qk1 - dpsum, 1e-24f)) + pls;
    const float mm  = fmaxf(lse, lr);
    const float log_norm = mm + logf(expf(lse - mm) + expf(lr - mm));
    const float pscale   = expf(pls - log_norm);
    Pscale[row] = pscale;
    for (int j = 0; j < 144; ++j) {
      int s = s_base + j;
      bool valid = (s >= 0) && (s < T_DIM) && (s >= t - HALF_W) && (s <= t + HALF_W - 1);
      Ab[row][j] = valid ? (expf(QKb[row][j] - log_norm) - DPb[row][j] * pscale) : 0.f;
    }
  }
  __syncthreads();

  // out = attn_local(16x144) @ V_band(144x64) + qkv * prime_scale
  for (int nt = 0; nt < 4; ++nt) {
    const int d = nt * 16 + n;
    v8f acc = {};
    for (int kk = 0; kk < 144 / 4; ++kk) {
      int k0  = kk * 4 + koff;
      int s0c = clampi(s_base + k0,     0, T_DIM - 1);
      int s1c = clampi(s_base + k0 + 1, 0, T_DIM - 1);
      v2f a, b;
      a.x = Ab[n][k0];  a.y = Ab[n][k0 + 1];
      b.x = vin[((size_t)s0c * H_DIM + h) * E_DIM + d];
      b.y = vin[((size_t)s1c * H_DIM + h) * E_DIM + d];
      acc = wmma4(a, b, acc);
    }
    for (int vv = 0; vv < 8; ++vv) {
      int row = vv + ((lane >> 4) << 3);
      int t   = qt * 16 + row;
      out[((size_t)t * H_DIM + h) * E_DIM + d] = acc[vv] + qkvAcc[nt][vv] * Pscale[row];
    }
  }
}

// ---------------------------------------------------------------------------
extern "C" void kernel_launch(void* const* d_in, const int* in_sizes, int n_in,
                              void* d_out, int out_size, void* d_ws, size_t ws_size,
                              hipStream_t stream)
{
  (void)in_sizes; (void)n_in; (void)out_size; (void)ws_size;
  const float* q    = (const float*)d_in[0];
  const float* k    = (const float*)d_in[1];
  const float* v    = (const float*)d_in[2];
  const float* proj = (const float*)d_in[3];
  float* out = (float*)d_out;
  float* ws  = (float*)d_ws;

  // workspace layout (floats); total ~10.9 MB
  size_t o = 0;
  float* qp      = ws + o; o += (size_t)H_DIM * T_DIM * F_DIM;   // 4 MB
  float* kp      = ws + o; o += (size_t)H_DIM * T_DIM * F_DIM;   // 4 MB
  float* kdash   = ws + o; o += (size_t)H_DIM * T_DIM * R_DIM;   // 2 MB
  float* kdiag   = ws + o; o += (size_t)H_DIM * T_DIM;
  float* krowmax = ws + o; o += (size_t)H_DIM * T_DIM;
  float* qls     = ws + o; o += (size_t)H_DIM * T_DIM;
  float* kls     = ws + o; o += (size_t)H_DIM;
  float* kvbuf   = ws + o; o += (size_t)H_DIM * F_DIM * E_DIM;
  float* ksumb   = ws + o; o += (size_t)H_DIM * F_DIM;

  featurize_kernel<<<2 * H_DIM * NTILES_T, 32, 0, stream>>>(
      q, k, proj, qp, qls, kdash, kdiag, krowmax);
  reduce_kls_kernel<<<H_DIM, 256, 0, stream>>>(krowmax, kls);
  kprime_kernel<<<(H_DIM * T_DIM * R_DIM) / 256, 256, 0, stream>>>(
      kdash, kdiag, kls, kp);
  kv_kernel<<<H_DIM * 32, 32, 0, stream>>>(v, kp, kvbuf);
  ksum_kernel<<<H_DIM, F_DIM, 0, stream>>>(kp, ksumb);
  sb_final_kernel<<<H_DIM * NTILES_T, 32, 0, stream>>>(
      q, k, v, qp, kp, qls, kls, kvbuf, ksumb, out);
}